// Encoder_12257836663202
// MI455X (gfx1250) — compile-verified
//
#include <hip/hip_runtime.h>

typedef __attribute__((ext_vector_type(2))) float v2f;
typedef __attribute__((ext_vector_type(8))) float v8f;

#define NEG_SLOPE 0.2f
#define LN_EPS 1e-5f

// ---------- helpers ----------
__device__ __forceinline__ unsigned f32_mono(float f) {
  unsigned b = __float_as_uint(f);
  return (b & 0x80000000u) ? ~b : (b | 0x80000000u);
}
__device__ __forceinline__ float mono_f32(unsigned u) {
  return (u & 0x80000000u) ? __uint_as_float(u & 0x7fffffffu) : __uint_as_float(~u);
}
__device__ __forceinline__ float warp_sum32(float v) {
#pragma unroll
  for (int m = 16; m >= 1; m >>= 1) v += __shfl_xor(v, m, 32);
  return v;
}

// ---------- WMMA GEMM: Out[M,128] = act(A[M,K] @ W[K,128] + bias) ----------
// one wave computes one 16x16 tile; V_WMMA_F32_16X16X4_F32, K stepped by 4.
__global__ void k_gemm_wmma(const float* __restrict__ A, int K,
                            const float* __restrict__ W,
                            const float* __restrict__ bias,
                            float* __restrict__ Out,
                            int Mtiles, int do_relu) {
  int wid  = threadIdx.x >> 5;
  int lane = threadIdx.x & 31;
  int tile = blockIdx.x * (blockDim.x >> 5) + wid;
  if (tile >= Mtiles * 8) return;          // uniform per wave -> EXEC all-ones inside
  int tm = tile >> 3, tn = tile & 7;
  int hi = lane >> 4;                      // 0: K=kb,kb+1   1: K=kb+2,kb+3
  int l15 = lane & 15;
  int arow = tm * 16 + l15;
  int bcol = tn * 16 + l15;
  const float* Arow = A + (size_t)arow * K;
  v8f c = {};
  for (int kb = 0; kb < K; kb += 4) {
    v2f a = *(const v2f*)(Arow + kb + 2 * hi);
    v2f b;
    b.x = W[(size_t)(kb + 2 * hi) * 128 + bcol];
    b.y = W[(size_t)(kb + 2 * hi + 1) * 128 + bcol];
    c = __builtin_amdgcn_wmma_f32_16x16x4_f32(false, a, false, b, (short)0, c,
                                              false, false);
  }
#pragma unroll
  for (int r = 0; r < 8; ++r) {
    int row = tm * 16 + r + 8 * hi;        // C/D: vgpr r -> M=r (lanes 0-15), M=r+8 (16-31)
    float v = c[r] + bias[bcol];
    if (do_relu) v = fmaxf(v, 0.0f);
    Out[(size_t)row * 128 + bcol] = v;
  }
}

// xs = concat(x[N,32], h[N,128]) @ W[160,128]  (no bias/act), concat done in the A-load
__global__ void k_gemm_concat_wmma(const float* __restrict__ Xin,
                                   const float* __restrict__ Hin,
                                   const float* __restrict__ W,
                                   float* __restrict__ Out, int Mtiles) {
  int wid  = threadIdx.x >> 5;
  int lane = threadIdx.x & 31;
  int tile = blockIdx.x * (blockDim.x >> 5) + wid;
  if (tile >= Mtiles * 8) return;
  int tm = tile >> 3, tn = tile & 7;
  int hi = lane >> 4;
  int l15 = lane & 15;
  int arow = tm * 16 + l15;
  int bcol = tn * 16 + l15;
  v8f c = {};
  for (int kb = 0; kb < 160; kb += 4) {    // kb<32 entirely x-side, kb>=32 entirely h-side
    v2f a;
    if (kb < 32) a = *(const v2f*)(Xin + (size_t)arow * 32 + kb + 2 * hi);
    else         a = *(const v2f*)(Hin + (size_t)arow * 128 + (kb - 32) + 2 * hi);
    v2f b;
    b.x = W[(size_t)(kb + 2 * hi) * 128 + bcol];
    b.y = W[(size_t)(kb + 2 * hi + 1) * 128 + bcol];
    c = __builtin_amdgcn_wmma_f32_16x16x4_f32(false, a, false, b, (short)0, c,
                                              false, false);
  }
#pragma unroll
  for (int r = 0; r < 8; ++r) {
    int row = tm * 16 + r + 8 * hi;
    Out[(size_t)row * 128 + bcol] = c[r];
  }
}

// ---------- small utility kernels ----------
__global__ void k_zero(float* p, int n) {
  int i = blockIdx.x * blockDim.x + threadIdx.x;
  if (i < n) p[i] = 0.0f;
}

// column sums of edge_attr [E,16] -> sum16 (block LDS reduce, then 16 atomics/block)
__global__ void k_colsum(const float* __restrict__ ea, float* __restrict__ sum16, int E) {
  __shared__ float sacc[16];
  int tid = threadIdx.x;
  if (tid < 16) sacc[tid] = 0.0f;
  __syncthreads();
  int c = tid & 15;
  int worker   = (blockIdx.x * blockDim.x + tid) >> 4;
  int nworkers = (gridDim.x * blockDim.x) >> 4;
  float s = 0.0f;
  for (int r = worker; r < E; r += nworkers) s += ea[(size_t)r * 16 + c];
  atomicAdd(&sacc[c], s);
  __syncthreads();
  if (tid < 16) atomicAdd(&sum16[tid], sacc[tid]);
}

// fold W_edge[l] with att_edge[l]:  w_eh[k*4+h] = sum_c W_edge[k,h*32+c]*att_edge[h,c]
// and a_e for self-loop rows: aeloop[h] = sum_k (mean_ea[k]) * w_eh[k,h]
__global__ void k_weh(const float* __restrict__ W_edge_l,
                      const float* __restrict__ att_edge_l,
                      const float* __restrict__ sum16,
                      float* __restrict__ w_eh, float* __restrict__ aeloop, int E) {
  int tid = threadIdx.x;   // blockDim == 64
  int k = tid >> 2, hh = tid & 3;
  float s = 0.0f;
  for (int c = 0; c < 32; ++c)
    s += W_edge_l[(size_t)k * 128 + hh * 32 + c] * att_edge_l[hh * 32 + c];
  w_eh[tid] = s;
  __syncthreads();
  if (tid < 4) {
    float inv = 1.0f / (float)E;
    float t = 0.0f;
    for (int kk = 0; kk < 16; ++kk) t += (sum16[kk] * inv) * w_eh[kk * 4 + tid];
    aeloop[tid] = t;
  }
}

// a_s[n,h], a_d[n,h] from xs
__global__ void k_as_ad(const float* __restrict__ xs,
                        const float* __restrict__ att_src_l,
                        const float* __restrict__ att_dst_l,
                        float* __restrict__ a_s, float* __restrict__ a_d, int N) {
  int idx = blockIdx.x * blockDim.x + threadIdx.x;
  if (idx >= N * 4) return;
  int n = idx >> 2, hh = idx & 3;
  const float* row = xs + (size_t)n * 128 + hh * 32;
  float s1 = 0.0f, s2 = 0.0f;
#pragma unroll
  for (int c = 0; c < 32; ++c) {
    float v = row[c];
    s1 += v * att_src_l[hh * 32 + c];
    s2 += v * att_dst_l[hh * 32 + c];
  }
  a_s[idx] = s1;
  a_d[idx] = s2;
}

__global__ void k_init_softmax(unsigned* __restrict__ keyb, float* __restrict__ den, int n) {
  int i = blockIdx.x * blockDim.x + threadIdx.x;
  if (i < n) { keyb[i] = 0x007FFFFFu; /* mono(-inf) */ den[i] = 0.0f; }
}

__global__ void k_init_acc(float* __restrict__ acc, const float* __restrict__ bias, int N) {
  int i = blockIdx.x * blockDim.x + threadIdx.x;
  if (i < N * 128) acc[i] = bias[i & 127];
}

// alpha = leaky_relu(a_s[src]+a_d[dst]+a_e); store alpha; atomicMax mono-key per (dst,h)
__global__ void k_alpha(const int* __restrict__ src, const int* __restrict__ dst,
                        const float* __restrict__ ea, const float* __restrict__ w_eh,
                        const float* __restrict__ aeloop,
                        const float* __restrict__ a_s, const float* __restrict__ a_d,
                        float* __restrict__ alpha_buf, unsigned* __restrict__ keyb,
                        int E, int E2) {
  int e = blockIdx.x * blockDim.x + threadIdx.x;
  if (e >= E2) return;
  int si, di;
  float ae[4];
  if (e < E) {
    si = src[e]; di = dst[e];
    float ar[16];
    const float4* ep = (const float4*)(ea + (size_t)e * 16);
#pragma unroll
    for (int q = 0; q < 4; ++q) {
      float4 v = ep[q];
      ar[q * 4 + 0] = v.x; ar[q * 4 + 1] = v.y; ar[q * 4 + 2] = v.z; ar[q * 4 + 3] = v.w;
    }
#pragma unroll
    for (int hh = 0; hh < 4; ++hh) {
      float s = 0.0f;
#pragma unroll
      for (int k = 0; k < 16; ++k) s += ar[k] * w_eh[k * 4 + hh];
      ae[hh] = s;
    }
  } else {
    si = e - E; di = si;
#pragma unroll
    for (int hh = 0; hh < 4; ++hh) ae[hh] = aeloop[hh];
  }
#pragma unroll
  for (int hh = 0; hh < 4; ++hh) {
    float a = a_s[(size_t)si * 4 + hh] + a_d[(size_t)di * 4 + hh] + ae[hh];
    a = (a > 0.0f) ? a : NEG_SLOPE * a;
    alpha_buf[(size_t)e * 4 + hh] = a;
    atomicMax(&keyb[(size_t)di * 4 + hh], f32_mono(a));
  }
}

// ex = exp(alpha - amax[dst]); in-place; atomicAdd denom[dst]
__global__ void k_exp(const int* __restrict__ dst, const unsigned* __restrict__ keyb,
                      float* __restrict__ exb, float* __restrict__ den, int E, int E2) {
  int e = blockIdx.x * blockDim.x + threadIdx.x;
  if (e >= E2) return;
  int di = (e < E) ? dst[e] : (e - E);
#pragma unroll
  for (int hh = 0; hh < 4; ++hh) {
    float am = mono_f32(keyb[(size_t)di * 4 + hh]);
    float v = __expf(exb[(size_t)e * 4 + hh] - am);
    exb[(size_t)e * 4 + hh] = v;
    atomicAdd(&den[(size_t)di * 4 + hh], v);
  }
}

// one wave per edge: acc[dst] += xs[src] * w;  lane owns 4 channels (one head)
__global__ void k_scatter(const int* __restrict__ src, const int* __restrict__ dst,
                          const float* __restrict__ xs, const float* __restrict__ exb,
                          const float* __restrict__ den, float* __restrict__ acc,
                          int E, int E2) {
  int e = blockIdx.x * (blockDim.x >> 5) + (threadIdx.x >> 5);
  if (e >= E2) return;
  int lane = threadIdx.x & 31;
  int si, di;
  if (e < E) { si = src[e]; di = dst[e]; } else { si = e - E; di = si; }
  int hh = lane >> 3;                                   // 4 channels * 8 lanes = 1 head
  float w = exb[(size_t)e * 4 + hh] / (den[(size_t)di * 4 + hh] + 1e-16f);
  float4 v = *(const float4*)(xs + (size_t)si * 128 + lane * 4);
  float* o = acc + (size_t)di * 128 + lane * 4;
  atomicAdd(o + 0, v.x * w);
  atomicAdd(o + 1, v.y * w);
  atomicAdd(o + 2, v.z * w);
  atomicAdd(o + 3, v.w * w);
}

// ---------- LayerNorm (one wave per row of 128) ----------
__device__ __forceinline__ float4 ln_row(const float* __restrict__ in, int c0,
                                         const float* __restrict__ gamma,
                                         const float* __restrict__ beta) {
  float4 v = *(const float4*)(in + c0);
  float mean = warp_sum32(v.x + v.y + v.z + v.w) * (1.0f / 128.0f);
  float dx = v.x - mean, dy = v.y - mean, dz = v.z - mean, dw = v.w - mean;
  float var = warp_sum32(dx * dx + dy * dy + dz * dz + dw * dw) * (1.0f / 128.0f);
  float inv = rsqrtf(var + LN_EPS);
  float4 o;
  o.x = dx * inv * gamma[c0 + 0] + beta[c0 + 0];
  o.y = dy * inv * gamma[c0 + 1] + beta[c0 + 1];
  o.z = dz * inv * gamma[c0 + 2] + beta[c0 + 2];
  o.w = dw * inv * gamma[c0 + 3] + beta[c0 + 3];
  return o;
}

__global__ void k_ln_elu(const float* __restrict__ acc, const float* __restrict__ gamma,
                         const float* __restrict__ beta, float* __restrict__ h, int N) {
  int n = blockIdx.x * (blockDim.x >> 5) + (threadIdx.x >> 5);
  if (n >= N) return;
  int c0 = (threadIdx.x & 31) * 4;
  float4 o = ln_row(acc + (size_t)n * 128, c0, gamma, beta);
  o.x = o.x > 0.0f ? o.x : __expf(o.x) - 1.0f;
  o.y = o.y > 0.0f ? o.y : __expf(o.y) - 1.0f;
  o.z = o.z > 0.0f ? o.z : __expf(o.z) - 1.0f;
  o.w = o.w > 0.0f ? o.w : __expf(o.w) - 1.0f;
  *(float4*)(h + (size_t)n * 128 + c0) = o;
}

__global__ void k_ln_store(const float* __restrict__ acc, const float* __restrict__ gamma,
                           const float* __restrict__ beta, float* __restrict__ out, int N) {
  int n = blockIdx.x * (blockDim.x >> 5) + (threadIdx.x >> 5);
  if (n >= N) return;
  int c0 = (threadIdx.x & 31) * 4;
  float4 o = ln_row(acc + (size_t)n * 128, c0, gamma, beta);
  *(float4*)(out + (size_t)n * 128 + c0) = o;
}

__global__ void k_ln_pool(const float* __restrict__ acc, const float* __restrict__ gamma,
                          const float* __restrict__ beta, const int* __restrict__ batch,
                          float* __restrict__ gsum, float* __restrict__ cnt, int N) {
  int n = blockIdx.x * (blockDim.x >> 5) + (threadIdx.x >> 5);
  if (n >= N) return;
  int lane = threadIdx.x & 31;
  int c0 = lane * 4;
  float4 o = ln_row(acc + (size_t)n * 128, c0, gamma, beta);
  int b = batch[n];
  float* gp = gsum + (size_t)b * 128 + c0;
  atomicAdd(gp + 0, o.x);
  atomicAdd(gp + 1, o.y);
  atomicAdd(gp + 2, o.z);
  atomicAdd(gp + 3, o.w);
  if (lane == 0) atomicAdd(&cnt[b], 1.0f);
}

__global__ void k_pool_div(const float* __restrict__ gsum, const float* __restrict__ cnt,
                           float* __restrict__ out, int n) {
  int i = blockIdx.x * blockDim.x + threadIdx.x;
  if (i < n) out[i] = gsum[i] / fmaxf(cnt[i >> 7], 1.0f);
}

// ---------- host ----------
extern "C" void kernel_launch(void* const* d_in, const int* in_sizes, int n_in,
                              void* d_out, int out_size, void* d_ws, size_t ws_size,
                              hipStream_t stream) {
  const float* x        = (const float*)d_in[0];
  const float* edge_attr= (const float*)d_in[1];
  const float* W_in     = (const float*)d_in[2];
  const float* b_in     = (const float*)d_in[3];
  const float* W_src    = (const float*)d_in[4];
  const float* W_edge   = (const float*)d_in[5];
  const float* att_src  = (const float*)d_in[6];
  const float* att_dst  = (const float*)d_in[7];
  const float* att_edge = (const float*)d_in[8];
  const float* gat_bias = (const float*)d_in[9];
  const float* ln_gamma = (const float*)d_in[10];
  const float* ln_beta  = (const float*)d_in[11];
  const float* W_out    = (const float*)d_in[12];
  const float* b_out    = (const float*)d_in[13];
  const float* ln_out_g = (const float*)d_in[14];
  const float* ln_out_b = (const float*)d_in[15];
  const float* W_g      = (const float*)d_in[16];
  const float* b_g      = (const float*)d_in[17];
  const float* g_gamma  = (const float*)d_in[18];
  const float* g_beta   = (const float*)d_in[19];
  const int*   eidx     = (const int*)d_in[20];
  const int*   batch    = (const int*)d_in[21];

  const int N  = in_sizes[0] / 32;
  const int E  = in_sizes[1] / 16;
  const int E2 = E + N;
  const int B  = out_size / 128 - N;
  const int Mtiles = N / 16;
  const int* srcp = eidx;
  const int* dstp = eidx + E;

  float* ws = (float*)d_ws;
  size_t oH   = 0;
  size_t oXS  = oH   + (size_t)N * 128;
  size_t oACC = oXS  + (size_t)N * 128;
  size_t oEX  = oACC + (size_t)N * 128;
  size_t oAS  = oEX  + (size_t)E2 * 4;
  size_t oAD  = oAS  + (size_t)N * 4;
  size_t oKEY = oAD  + (size_t)N * 4;
  size_t oDEN = oKEY + (size_t)N * 4;
  size_t oWEH = oDEN + (size_t)N * 4;
  size_t oSUM = oWEH + 64;
  size_t oAEL = oSUM + 16;
  size_t oGS  = oAEL + 16;
  size_t oCNT = oGS  + (size_t)B * 128;

  float*    hbuf   = ws + oH;
  float*    xs     = ws + oXS;
  float*    acc    = ws + oACC;
  float*    exb    = ws + oEX;
  float*    a_s    = ws + oAS;
  float*    a_d    = ws + oAD;
  unsigned* keyb   = (unsigned*)(ws + oKEY);
  float*    den    = ws + oDEN;
  float*    weh    = ws + oWEH;
  float*    sum16  = ws + oSUM;
  float*    aeloop = ws + oAEL;
  float*    gsum   = ws + oGS;
  float*    cnt    = ws + oCNT;

  const int gemm_blocks = (Mtiles * 8 + 3) / 4;   // 4 waves / 128-thread block

  // h = relu(x @ W_in + b_in)
  k_gemm_wmma<<<gemm_blocks, 128, 0, stream>>>(x, 32, W_in, b_in, hbuf, Mtiles, 1);

  // column sums of edge_attr (for self-loop edge features)
  k_zero<<<1, 64, 0, stream>>>(sum16, 16);
  k_colsum<<<256, 256, 0, stream>>>(edge_attr, sum16, E);

  for (int l = 0; l < 2; ++l) {
    k_weh<<<1, 64, 0, stream>>>(W_edge + (size_t)l * 16 * 128,
                                att_edge + (size_t)l * 128, sum16, weh, aeloop, E);
    k_gemm_concat_wmma<<<gemm_blocks, 128, 0, stream>>>(
        x, hbuf, W_src + (size_t)l * 160 * 128, xs, Mtiles);
    k_as_ad<<<(N * 4 + 255) / 256, 256, 0, stream>>>(
        xs, att_src + (size_t)l * 128, att_dst + (size_t)l * 128, a_s, a_d, N);
    k_init_softmax<<<(N * 4 + 255) / 256, 256, 0, stream>>>(keyb, den, N * 4);
    k_init_acc<<<(N * 128 + 255) / 256, 256, 0, stream>>>(acc, gat_bias + (size_t)l * 128, N);
    k_alpha<<<(E2 + 255) / 256, 256, 0, stream>>>(srcp, dstp, edge_attr, weh, aeloop,
                                                  a_s, a_d, exb, keyb, E, E2);
    k_exp<<<(E2 + 255) / 256, 256, 0, stream>>>(dstp, keyb, exb, den, E, E2);
    k_scatter<<<(E2 + 7) / 8, 256, 0, stream>>>(srcp, dstp, xs, exb, den, acc, E, E2);
    k_ln_elu<<<(N + 7) / 8, 256, 0, stream>>>(acc, ln_gamma + (size_t)l * 128,
                                              ln_beta + (size_t)l * 128, hbuf, N);
  }

  float* out_nodes = (float*)d_out;
  float* out_graph = out_nodes + (size_t)N * 128;

  // node_embeddings = LN(relu(h@W_out+b_out))
  k_gemm_wmma<<<gemm_blocks, 128, 0, stream>>>(hbuf, 128, W_out, b_out, acc, Mtiles, 1);
  k_ln_store<<<(N + 7) / 8, 256, 0, stream>>>(acc, ln_out_g, ln_out_b, out_nodes, N);

  // graph_embeddings = mean_b LN(relu(h@W_g+b_g))
  k_gemm_wmma<<<gemm_blocks, 128, 0, stream>>>(hbuf, 128, W_g, b_g, acc, Mtiles, 1);
  k_zero<<<(B * 128 + B + 255) / 256, 256, 0, stream>>>(gsum, B * 128 + B); // gsum + cnt
  k_ln_pool<<<(N + 7) / 8, 256, 0, stream>>>(acc, g_gamma, g_beta, batch, gsum, cnt, N);
  k_pool_div<<<(B * 128 + 255) / 256, 256, 0, stream>>>(gsum, cnt, out_graph, B * 128);
}